// GraphNet_54735063220776
// MI455X (gfx1250) — compile-verified
//
#include <hip/hip_runtime.h>
#include <math.h>

// ---------------------------------------------------------------------------
// GCN forward for MI455X (gfx1250, wave32).
//  - f32 WMMA (V_WMMA_F32_16X16X4_F32) for all three GEMMs (exact fp32 math)
//  - 64-row blocking: W staged to LDS once per 64 output rows; each wave holds
//    4 accumulator tiles and reuses one B register pair across them per k-step
//  - L2-resident edge scatter with float atomics (h is 32MB << 192MB L2)
//  - self-loop + bias folded into aggregation-buffer seeding (saves one pass)
// ---------------------------------------------------------------------------

typedef __attribute__((ext_vector_type(2))) float v2f;
typedef __attribute__((ext_vector_type(4))) float f4;
typedef __attribute__((ext_vector_type(8))) float v8f;

// ------------------------------ small kernels ------------------------------

__global__ void k_fill(float* __restrict__ p, float v, int n) {
    int i = blockIdx.x * blockDim.x + threadIdx.x;
    if (i < n) p[i] = v;
}

__global__ void k_deg_scatter(const int* __restrict__ dst,
                              const float* __restrict__ ew,
                              float* __restrict__ deg, int E) {
    int e = blockIdx.x * blockDim.x + threadIdx.x;
    if (e < E) atomicAdd(&deg[dst[e]], ew[e]);
}

__global__ void k_rsqrt_inplace(float* __restrict__ d, int n) {
    int i = blockIdx.x * blockDim.x + threadIdx.x;
    if (i < n) d[i] = rsqrtf(d[i]);
}

__global__ void k_edge_norm(const int* __restrict__ src,
                            const int* __restrict__ dst,
                            const float* __restrict__ ew,
                            const float* __restrict__ dinv,
                            float* __restrict__ nrm, int E) {
    int e = blockIdx.x * blockDim.x + threadIdx.x;
    if (e < E) nrm[e] = dinv[src[e]] * ew[e] * dinv[dst[e]];
}

// Seed aggregation buffer with self-loop + bias: agg = h * dinv^2 + bias.
__global__ void k_seed_agg(const float* __restrict__ h,
                           const float* __restrict__ dinv,
                           const float* __restrict__ bias,
                           float* __restrict__ agg, int n32) {
    int i = blockIdx.x * blockDim.x + threadIdx.x;
    if (i >= n32) return;
    int row = i >> 5, c4 = i & 31;
    float s = dinv[row];
    s *= s;
    f4 hv = *(const f4*)&h[(size_t)i * 4];
    f4 bv = *(const f4*)&bias[c4 * 4];
    f4 r;
    r.x = fmaf(hv.x, s, bv.x);
    r.y = fmaf(hv.y, s, bv.y);
    r.z = fmaf(hv.z, s, bv.z);
    r.w = fmaf(hv.w, s, bv.w);
    *(f4*)&agg[(size_t)i * 4] = r;
}

// One wave32 per edge: lane L moves features [4L, 4L+4) of a 128-wide row.
__launch_bounds__(256)
__global__ void k_edge_scatter(const float* __restrict__ h,
                               const int* __restrict__ src,
                               const int* __restrict__ dst,
                               const float* __restrict__ nrm,
                               float* __restrict__ agg, int E) {
    int e = blockIdx.x * 8 + (threadIdx.x >> 5);
    if (e >= E) return;
    int lane = threadIdx.x & 31;
    int s = src[e], d = dst[e];
    float w = nrm[e];
    f4 v = *(const f4*)&h[(size_t)s * 128 + lane * 4];
    float* ap = &agg[(size_t)d * 128 + lane * 4];
    atomicAdd(ap + 0, v.x * w);
    atomicAdd(ap + 1, v.y * w);
    atomicAdd(ap + 2, v.z * w);
    atomicAdd(ap + 3, v.w * w);
}

// out = relu(in), float4-vectorized.
__global__ void k_relu_copy(const float* __restrict__ in,
                            float* __restrict__ out, int n4) {
    int i = blockIdx.x * blockDim.x + threadIdx.x;
    if (i >= n4) return;
    f4 v = *(const f4*)&in[(size_t)i * 4];
    f4 r;
    r.x = fmaxf(v.x, 0.0f);
    r.y = fmaxf(v.y, 0.0f);
    r.z = fmaxf(v.z, 0.0f);
    r.w = fmaxf(v.w, 0.0f);
    *(f4*)&out[(size_t)i * 4] = r;
}

// ------------------------------- WMMA GEMM ---------------------------------
// Y[64 x NCOLS] tile per block:  Y = X(64x128) * W(128xNCOLS)  (+ bias)
// Block = NCOLS/16 waves; wave w owns column strip [16w, 16w+16) and keeps
// 4 accumulator tiles (rows t*16..t*16+15).  Per k-step one B register pair
// is shared by 4 WMMAs -> 4x less LDS B traffic, 4x W-staging amortization.
template <int NCOLS, bool HAS_BIAS>
__launch_bounds__(NCOLS * 2)
__global__ void gemm_k128_wmma(const float* __restrict__ X,
                               const float* __restrict__ W,
                               const float* __restrict__ bias,
                               float* __restrict__ Y) {
    constexpr int ROWS = 64;        // rows per block (4 tiles of 16)
    constexpr int XS = 132;         // X LDS stride: banks = (4*row + k) % 64
    constexpr int WSID = NCOLS + 8; // W LDS stride: half-waves 16 banks apart
    __shared__ __align__(16) float sX[ROWS * XS];
    __shared__ __align__(16) float sW[128 * WSID];

    const int tid = threadIdx.x;
    constexpr int NTHR = NCOLS * 2;
    const int rowBase = blockIdx.x * ROWS;

    // Stage X tile (64 x 128) via float4.
    {
        const f4* Xg = (const f4*)(X + (size_t)rowBase * 128);
        for (int q = tid; q < ROWS * 32; q += NTHR) {
            int r = q >> 5, c4 = q & 31;
            *(f4*)&sX[r * XS + c4 * 4] = Xg[r * 32 + c4];
        }
    }
    // Stage W (128 x NCOLS) via float4.
    {
        constexpr int C4 = NCOLS / 4;
        const f4* Wg = (const f4*)W;
        for (int q = tid; q < 128 * C4; q += NTHR) {
            int r = q / C4, c4 = q % C4;
            *(f4*)&sW[r * WSID + c4 * 4] = Wg[q];
        }
    }
    __syncthreads();

    const int lane = tid & 31;
    const int wave = tid >> 5;
    const int colBase = wave * 16;
    const int mr = lane & 15;          // A row / B,D col within 16x16 tile
    const int kh = (lane >> 4) * 2;    // K sub-offset for this half-wave

    const float* ax = &sX[mr * XS + kh];
    const float* bw = &sW[kh * WSID + colBase + mr];

    v8f acc[4] = {{}, {}, {}, {}};
#pragma unroll
    for (int k0 = 0; k0 < 128; k0 += 4) {
        v2f b;
        b.x = bw[k0 * WSID];
        b.y = bw[k0 * WSID + WSID];
#pragma unroll
        for (int t = 0; t < 4; ++t) {
            v2f a;
            a.x = ax[t * 16 * XS + k0];
            a.y = ax[t * 16 * XS + k0 + 1];
            acc[t] = __builtin_amdgcn_wmma_f32_16x16x4_f32(
                /*neg_a=*/false, a, /*neg_b=*/false, b,
                /*c_mod=*/(short)0, acc[t], /*reuse_a=*/false, /*reuse_b=*/false);
        }
    }

    // D layout: VGPR i -> M = i + (lane>=16 ? 8 : 0), N = lane%16.
    const int mBase = (lane >> 4) * 8;
    float bv = 0.0f;
    if (HAS_BIAS) bv = bias[colBase + mr];
#pragma unroll
    for (int t = 0; t < 4; ++t) {
        float* Yt = Y + (size_t)(rowBase + t * 16) * NCOLS + colBase + mr;
#pragma unroll
        for (int i = 0; i < 8; ++i) {
            Yt[(size_t)(mBase + i) * NCOLS] = acc[t][i] + bv;
        }
    }
}

// ------------------------------- launcher ----------------------------------

static inline int ceil_div(int a, int b) { return (a + b - 1) / b; }

extern "C" void kernel_launch(void* const* d_in, const int* in_sizes, int n_in,
                              void* d_out, int out_size, void* d_ws, size_t ws_size,
                              hipStream_t stream) {
    const float* x   = (const float*)d_in[0];
    const int*   ei  = (const int*)d_in[1];
    const float* ew  = (const float*)d_in[2];
    // d_in[3] = batch (unused: graphs are contiguous 1024-node blocks)
    const float* W1  = (const float*)d_in[4];
    const float* b1  = (const float*)d_in[5];
    const float* W2  = (const float*)d_in[6];
    const float* b2  = (const float*)d_in[7];
    const float* fcW = (const float*)d_in[8];
    const float* fcb = (const float*)d_in[9];
    float* out = (float*)d_out;

    const int N = in_sizes[0] / 128;   // 65536
    const int E = in_sizes[2];         // 1048576
    const int* src = ei;
    const int* dst = ei + E;

    // Workspace layout (floats): H[N*128] | A[N*128] | D[N] | NRM[E]  (~71 MB)
    float* H   = (float*)d_ws;
    float* A   = H + (size_t)N * 128;
    float* D   = A + (size_t)N * 128;
    float* NRM = D + N;

    const int BT = 256;

    // Degrees + symmetric normalization factors (shared by both layers).
    k_fill<<<ceil_div(N, BT), BT, 0, stream>>>(D, 1.0f, N);
    k_deg_scatter<<<ceil_div(E, BT), BT, 0, stream>>>(dst, ew, D, E);
    k_rsqrt_inplace<<<ceil_div(N, BT), BT, 0, stream>>>(D, N);
    k_edge_norm<<<ceil_div(E, BT), BT, 0, stream>>>(src, dst, ew, D, NRM, E);

    // Layer 1: H = x@W1 ; A = H*dinv^2 + b1 ; A += scatter(H) ; H = relu(A)
    gemm_k128_wmma<128, false><<<N / 64, 256, 0, stream>>>(x, W1, nullptr, H);
    k_seed_agg<<<ceil_div(N * 32, BT), BT, 0, stream>>>(H, D, b1, A, N * 32);
    k_edge_scatter<<<ceil_div(E, 8), 256, 0, stream>>>(H, src, dst, NRM, A, E);
    k_relu_copy<<<ceil_div(N * 32, BT), BT, 0, stream>>>(A, H, N * 32);

    // Layer 2: A = H@W2 ; H = A*dinv^2 + b2 ; H += scatter(A) ; H = relu(H)
    gemm_k128_wmma<128, false><<<N / 64, 256, 0, stream>>>(H, W2, nullptr, A);
    k_seed_agg<<<ceil_div(N * 32, BT), BT, 0, stream>>>(A, D, b2, H, N * 32);
    k_edge_scatter<<<ceil_div(E, 8), 256, 0, stream>>>(A, src, dst, NRM, H, E);
    k_relu_copy<<<ceil_div(N * 32, BT), BT, 0, stream>>>(H, H, N * 32);

    // FC head: out = H @ fcW + fcb   ([65536 x 64])
    gemm_k128_wmma<64, true><<<N / 64, 128, 0, stream>>>(H, fcW, fcb, out);
}